// WESUP_18133351924316
// MI455X (gfx1250) — compile-verified
//
#include <hip/hip_runtime.h>
#include <hip/hip_bf16.h>

#define C_CH   4224
#define HW     50176
#define S_SEG  100
#define H1DIM  1024
#define D3DIM  32
#define M_ROWS 100
#define SIM_TH 0.8f

typedef __attribute__((ext_vector_type(2))) float v2f;
typedef __attribute__((ext_vector_type(8))) float v8f;

// ---------------- segment counts + label counts (1 block) ----------------
__global__ void seg_cnt_kernel(const int* __restrict__ sp, const int* __restrict__ y,
                               float* __restrict__ cnt, float* __restrict__ lcnt) {
  __shared__ int cb[S_SEG];
  __shared__ int lb[S_SEG * 3];
  int t = threadIdx.x;
  for (int i = t; i < S_SEG; i += blockDim.x) cb[i] = 0;
  for (int i = t; i < S_SEG * 3; i += blockDim.x) lb[i] = 0;
  __syncthreads();
  for (int p = t; p < HW; p += blockDim.x) {
    int s = sp[p];
    atomicAdd(&cb[s], 1);
    atomicAdd(&lb[s * 3 + y[p]], 1);
  }
  __syncthreads();
  for (int i = t; i < S_SEG; i += blockDim.x) cnt[i] = (float)cb[i];
  for (int i = t; i < S_SEG * 3; i += blockDim.x) lcnt[i] = (float)lb[i];
}

// ---------------- streaming segment sum: one block per channel ----------------
// Reads the channel row coalesced (200KB), accumulates into per-wave LDS bins
// (8 waves x 100 bins) via ds_add_f32 atomics, then writes feat[s*C + c].
__global__ void seg_sum_kernel(const float* __restrict__ fm, const int* __restrict__ sp,
                               float* __restrict__ feat) {
  __shared__ float bins[8 * S_SEG];
  int t = threadIdx.x;
  int w = t >> 5;  // wave32
  int c = blockIdx.x;
  for (int i = t; i < 8 * S_SEG; i += 256) bins[i] = 0.f;
  __syncthreads();
  const float* row = fm + (size_t)c * HW;
  for (int p = t; p < HW; p += 256) {
    atomicAdd(&bins[w * S_SEG + sp[p]], row[p]);
  }
  __syncthreads();
  for (int s = t; s < S_SEG; s += 256) {
    float acc = 0.f;
#pragma unroll
    for (int ww = 0; ww < 8; ww++) acc += bins[ww * S_SEG + s];
    feat[(size_t)s * C_CH + c] = acc;
  }
}

// ---------------- feat = sums / max(cnt,1), row norms ----------------
__global__ void feat_norm_kernel(float* __restrict__ feat, const float* __restrict__ cnt,
                                 float* __restrict__ nrm) {
  __shared__ float red[256];
  int s = blockIdx.x, t = threadIdx.x;
  float inv = 1.f / fmaxf(cnt[s], 1.f);
  float acc = 0.f;
  float* row = feat + (size_t)s * C_CH;
  for (int c = t; c < C_CH; c += 256) {
    float v = row[c] * inv;
    row[c] = v;
    acc += v * v;
  }
  red[t] = acc;
  __syncthreads();
  for (int off = 128; off > 0; off >>= 1) {
    if (t < off) red[t] += red[t + off];
    __syncthreads();
  }
  if (t == 0) nrm[s] = sqrtf(red[0]);
}

// ---------------- per-segment label logic ----------------
__global__ void labels_kernel(const float* __restrict__ cnt, const float* __restrict__ lcnt,
                              int* __restrict__ labels) {
  int t = threadIdx.x + blockIdx.x * blockDim.x;
  if (t >= S_SEG) return;
  float c0 = lcnt[t * 3], c1 = lcnt[t * 3 + 1], c2 = lcnt[t * 3 + 2];
  int lab = (c0 >= c1 && c0 >= c2) ? 0 : ((c1 >= c2) ? 1 : 2);   // argmax, first-max wins
  int ovr = (c2 > floorf(cnt[t] * 0.5f)) ? 2 : ((c1 >= 1.f) ? 1 : 0);
  labels[t] = (lab == 0) ? ovr : lab;
}

// ---------------- affinity row + propagation: one block per segment i ----------------
__global__ void prop_kernel(const float* __restrict__ feat, const float* __restrict__ nrm,
                            const int* __restrict__ labels, float* __restrict__ out_prop) {
  __shared__ float vals[S_SEG];
  int i = blockIdx.x, t = threadIdx.x;
  const float NEG_INF = -__builtin_huge_valf();
  if (t < S_SEG) {
    float v = NEG_INF;
    if (labels[t] != 0) {
      const float* fi = feat + (size_t)i * C_CH;
      const float* fj = feat + (size_t)t * C_CH;
      float dot = 0.f;
      for (int c = 0; c < C_CH; c++) dot += fi[c] * fj[c];
      v = dot / (nrm[i] * nrm[t]);
    }
    vals[t] = v;
  }
  __syncthreads();
  if (t == 0) {
    float best = NEG_INF;
    int bidx = 0;
    for (int j = 0; j < S_SEG; j++) {
      if (vals[j] > best) { best = vals[j]; bidx = j; }  // strict > == argmax first-max
    }
    int li = labels[i];
    int prop = (li == 0 && best >= SIM_TH) ? labels[bidx] : li;
    out_prop[i] = (float)prop;
  }
}

// ---------------- fp32 WMMA GEMM + bias + relu ----------------
// One wave32 per 16x16 tile of Out = relu(A(MxK) * B(KxN) + bias).
// V_WMMA_F32_16X16X4_F32 fragment layout (ISA 7.12.2):
//   A 16x4 : lane l holds row (l&15), K = {0,1} (lanes 0-15) / {2,3} (lanes 16-31)
//   B 4x16 : lane l holds col (l&15), same K split
//   C/D    : acc[r] -> row r + 8*(l>=16), col (l&15)
__global__ void gemm_relu_wmma(const float* __restrict__ A, const float* __restrict__ B,
                               const float* __restrict__ bias, float* __restrict__ Out,
                               int M, int N, int K) {
  int lane = threadIdx.x;      // 32 threads = 1 wave, EXEC all ones
  int hi = lane >> 4;          // 0/1
  int lo = lane & 15;
  int nbase = blockIdx.x * 16;
  int mbase = blockIdx.y * 16;
  int arow = mbase + lo;
  float amask = (arow < M) ? 1.f : 0.f;      // mask via multiply: no EXEC divergence
  int arowc = (arow < M) ? arow : 0;         // clamped address: always a legal load
  int col = nbase + lo;

  const float* Arow = A + (size_t)arowc * K + 2 * hi;
  const float* Bcol = B + col + (size_t)(2 * hi) * N;

  v8f acc = {};
#pragma unroll 2
  for (int k = 0; k < K; k += 4) {
    v2f a;
    a.x = Arow[k] * amask;
    a.y = Arow[k + 1] * amask;
    v2f b;
    b.x = Bcol[(size_t)k * N];
    b.y = Bcol[(size_t)k * N + N];
    acc = __builtin_amdgcn_wmma_f32_16x16x4_f32(false, a, false, b, (short)0, acc,
                                                false, false);
  }

  float bv = bias[col];
#pragma unroll
  for (int r = 0; r < 8; r++) {
    int row = mbase + hi * 8 + r;
    if (row < M) {
      float v = acc[r] + bv;
      Out[(size_t)row * N + col] = v > 0.f ? v : 0.f;
    }
  }
}

// ---------------- final 32->2 classifier + softmax + argmax ----------------
__global__ void classifier_kernel(const float* __restrict__ h3, const float* __restrict__ Wc,
                                  const float* __restrict__ bc, float* __restrict__ out) {
  int t = threadIdx.x + blockIdx.x * blockDim.x;
  if (t >= S_SEG) return;
  float l0 = bc[0], l1 = bc[1];
  const float* hr = h3 + t * D3DIM;
#pragma unroll
  for (int d = 0; d < D3DIM; d++) {
    l0 += hr[d] * Wc[d * 2 + 0];
    l1 += hr[d] * Wc[d * 2 + 1];
  }
  float m = fmaxf(l0, l1);
  float e0 = expf(l0 - m), e1 = expf(l1 - m);
  float inv = 1.f / (e0 + e1);
  out[2 * t + 0] = e0 * inv;
  out[2 * t + 1] = e1 * inv;
  out[200 + t] = (l1 > l0) ? 1.f : 0.f;   // argmax; tie -> 0 (matches jnp.argmax)
}

extern "C" void kernel_launch(void* const* d_in, const int* in_sizes, int n_in,
                              void* d_out, int out_size, void* d_ws, size_t ws_size,
                              hipStream_t stream) {
  const float* fm = (const float*)d_in[0];
  const int*   sp = (const int*)d_in[1];
  const int*   y  = (const int*)d_in[2];
  const float* W1 = (const float*)d_in[3];
  const float* b1 = (const float*)d_in[4];
  const float* W2 = (const float*)d_in[5];
  const float* b2 = (const float*)d_in[6];
  const float* W3 = (const float*)d_in[7];
  const float* b3 = (const float*)d_in[8];
  const float* Wc = (const float*)d_in[9];
  const float* bc = (const float*)d_in[10];

  float* out = (float*)d_out;
  float* ws  = (float*)d_ws;

  // workspace layout (floats): total 631,000 floats (~2.5 MB)
  float* feat = ws;                       // 100*4224 = 422400 (sums, then feat in-place)
  float* h1   = feat + 422400;            // 100*1024 padded usage, 102400
  float* h2   = h1 + 102400;              // 102400
  float* h3   = h2 + 102400;              // 100*32 = 3200
  float* cnt  = h3 + 3200;                // 100
  float* lcnt = cnt + 100;                // 300
  float* nrm  = lcnt + 300;               // 100
  int*   labels = (int*)(nrm + 100);      // 100

  seg_cnt_kernel<<<1, 1024, 0, stream>>>(sp, y, cnt, lcnt);
  seg_sum_kernel<<<C_CH, 256, 0, stream>>>(fm, sp, feat);
  labels_kernel<<<1, 128, 0, stream>>>(cnt, lcnt, labels);
  feat_norm_kernel<<<S_SEG, 256, 0, stream>>>(feat, cnt, nrm);   // feat = sums/cnt, norms
  prop_kernel<<<S_SEG, 128, 0, stream>>>(feat, nrm, labels, out + 300);

  gemm_relu_wmma<<<dim3(H1DIM / 16, 7), 32, 0, stream>>>(feat, W1, b1, h1, M_ROWS, H1DIM, C_CH);
  gemm_relu_wmma<<<dim3(H1DIM / 16, 7), 32, 0, stream>>>(h1, W2, b2, h2, M_ROWS, H1DIM, H1DIM);
  gemm_relu_wmma<<<dim3(D3DIM / 16, 7), 32, 0, stream>>>(h2, W3, b3, h3, M_ROWS, D3DIM, H1DIM);
  classifier_kernel<<<1, 128, 0, stream>>>(h3, Wc, bc, out);
}